// RippleNet_23502061043982
// MI455X (gfx1250) — compile-verified
//
#include <hip/hip_runtime.h>
#include <hip/hip_bf16.h>

typedef __attribute__((ext_vector_type(2))) float v2f;
typedef __attribute__((ext_vector_type(8))) float v8f;

#define N_ENTITY 100000
#define N_RELATION 50
#define DIM 64
#define N_HOP 2
#define N_MEMORY 32
#define BATCH 512
#define KGE_WEIGHT 0.01f
#define L2_WEIGHT 1e-7f

// ---------------------------------------------------------------------------
// Kernel 1: u_emb[b,d] = sigmoid(mean over 32 hop-0 head embeddings)
// Also zero the global accumulators (block 0) before the atomic kernels run.
// ---------------------------------------------------------------------------
__global__ __launch_bounds__(64) void ripple_uemb(
    const float* __restrict__ ent, const int* __restrict__ mh0,
    float* __restrict__ U, float* __restrict__ accum) {
  const int b = blockIdx.x;
  const int d = threadIdx.x;
  if (b == 0 && d < 2) accum[d] = 0.0f;
  const int* row = mh0 + b * N_MEMORY;
  float s = 0.0f;
#pragma unroll 8
  for (int j = 0; j < N_MEMORY; ++j)
    s += ent[(size_t)row[j] * DIM + d];
  s *= (1.0f / (float)N_MEMORY);
  U[b * DIM + d] = 1.0f / (1.0f + __expf(-s));
}

// ---------------------------------------------------------------------------
// Kernel 2: scores = U @ W_out^T + b_out via V_WMMA_F32_16X16X4_F32.
// One wave -> 16(M) x 32(N) tile, 16 k-steps of K=4, 2 accumulators.
// A layout (16x4 f32): lanes0-15 = M rows; VGPR0 holds K0(lo half)/K2(hi half),
// VGPR1 holds K1/K3 -> a single float2 load per lane per k-step.
// B[k][n] = W_out[n][k]: same float2 pattern along each W_out row.
// ---------------------------------------------------------------------------
__global__ __launch_bounds__(128) void ripple_scores_wmma(
    const float* __restrict__ U, const float* __restrict__ W,
    const float* __restrict__ bias, float* __restrict__ out) {
  const int lane = threadIdx.x & 31;
  const int wave = threadIdx.x >> 5;
  const int tile = blockIdx.x * 4 + wave;   // 0 .. 99999 exactly
  const int mtile = tile & 31;              // 32 M-tiles
  const int ntile = tile >> 5;              // 3125 N-tiles of 32
  const int half = lane >> 4;               // 0: K0/K1, 1: K2/K3
  const int l16  = lane & 15;

  const float* a_ptr  = U + (size_t)(mtile * 16 + l16) * DIM + half * 2;
  const int n0 = ntile * 32 + l16;
  const int n1 = n0 + 16;
  const float* b_ptr0 = W + (size_t)n0 * DIM + half * 2;
  const float* b_ptr1 = W + (size_t)n1 * DIM + half * 2;

  const float bias0 = bias[n0];
  const float bias1 = bias[n1];
  v8f c0, c1;
#pragma unroll
  for (int j = 0; j < 8; ++j) { c0[j] = bias0; c1[j] = bias1; }

#pragma unroll
  for (int k = 0; k < 16; ++k) {
    v2f a  = *(const v2f*)(a_ptr  + 4 * k);
    v2f b0 = *(const v2f*)(b_ptr0 + 4 * k);
    v2f b1 = *(const v2f*)(b_ptr1 + 4 * k);
    c0 = __builtin_amdgcn_wmma_f32_16x16x4_f32(false, a, false, b0,
                                               (short)0, c0, false, false);
    c1 = __builtin_amdgcn_wmma_f32_16x16x4_f32(false, a, false, b1,
                                               (short)0, c1, false, false);
  }

  // C layout: VGPR j -> row (j + 8*half), col = l16
  const int mrow = mtile * 16 + half * 8;
  float* o = out + (size_t)mrow * N_ENTITY + (size_t)ntile * 32 + l16;
#pragma unroll
  for (int j = 0; j < 8; ++j) {
    o[(size_t)j * N_ENTITY]      = c0[j];
    o[(size_t)j * N_ENTITY + 16] = c1[j];
  }
}

// ---------------------------------------------------------------------------
// Kernel 3: per-batch  lse(scores[b,:]) - scores[b, items[b]]  (online logsumexp)
// ---------------------------------------------------------------------------
__global__ __launch_bounds__(256) void ripple_lse(
    const float* __restrict__ scores, const int* __restrict__ items,
    float* __restrict__ base_arr) {
  const int b = blockIdx.x;
  const int tid = threadIdx.x;
  const float* row = scores + (size_t)b * N_ENTITY;
  float m = -3.402823466e38f, s = 0.0f;
  for (int i = tid; i < N_ENTITY; i += 256) {
    float x = row[i];
    float nm = fmaxf(m, x);
    s = s * __expf(m - nm) + __expf(x - nm);
    m = nm;
  }
  __shared__ float sm[256];
  __shared__ float ss[256];
  sm[tid] = m; ss[tid] = s;
  __syncthreads();
  for (int off = 128; off > 0; off >>= 1) {
    if (tid < off) {
      float m2 = sm[tid + off], s2 = ss[tid + off];
      float nm = fmaxf(sm[tid], m2);
      ss[tid] = ss[tid] * __expf(sm[tid] - nm) + s2 * __expf(m2 - nm);
      sm[tid] = nm;
    }
    __syncthreads();
  }
  if (tid == 0)
    base_arr[b] = sm[0] + __logf(ss[0]) - row[items[b]];
}

// ---------------------------------------------------------------------------
// Kernel 4: one wave per (hop,b,m) triple: q = h^T R t, accumulate
// sigmoid(q) and the L2 term (h^2 + t^2 + R^2) into global accumulators.
// ---------------------------------------------------------------------------
__global__ __launch_bounds__(256) void ripple_hrt(
    const float* __restrict__ ent, const float* __restrict__ rel,
    const int* __restrict__ mh, const int* __restrict__ mr,
    const int* __restrict__ mt, float* __restrict__ accum) {
  const int lane = threadIdx.x & 31;
  const int wid = blockIdx.x * 8 + (threadIdx.x >> 5);  // 0..32767

  const int hid = mh[wid];
  const int rid = mr[wid];
  const int tid2 = mt[wid];
  const float* h = ent + (size_t)hid * DIM;
  const float* t = ent + (size_t)tid2 * DIM;
  const float* R = rel + (size_t)rid * (DIM * DIM);

  const float h0 = h[lane], h1 = h[lane + 32];
  const float t0 = t[lane], t1 = t[lane + 32];

  float a0 = 0.0f, a1 = 0.0f, rsq = 0.0f;
#pragma unroll 4
  for (int d = 0; d < 32; ++d) {
    float hd = __shfl(h0, d, 32);
    float r0 = R[d * DIM + lane];
    float r1 = R[d * DIM + lane + 32];
    a0 = fmaf(hd, r0, a0); a1 = fmaf(hd, r1, a1);
    rsq = fmaf(r0, r0, rsq); rsq = fmaf(r1, r1, rsq);
  }
#pragma unroll 4
  for (int d = 0; d < 32; ++d) {
    float hd = __shfl(h1, d, 32);
    float r0 = R[(d + 32) * DIM + lane];
    float r1 = R[(d + 32) * DIM + lane + 32];
    a0 = fmaf(hd, r0, a0); a1 = fmaf(hd, r1, a1);
    rsq = fmaf(r0, r0, rsq); rsq = fmaf(r1, r1, rsq);
  }

  float part = a0 * t0 + a1 * t1;                          // partial of q
  float l2p = h0 * h0 + h1 * h1 + t0 * t0 + t1 * t1 + rsq; // partial of L2
#pragma unroll
  for (int off = 16; off > 0; off >>= 1) {
    part += __shfl_xor(part, off, 32);
    l2p  += __shfl_xor(l2p,  off, 32);
  }
  if (lane == 0) {
    atomicAdd(accum + 0, 1.0f / (1.0f + __expf(-part)));
    atomicAdd(accum + 1, l2p);
  }
}

// ---------------------------------------------------------------------------
// Kernel 5: combine scalars -> d_out[0..3]
// ---------------------------------------------------------------------------
__global__ __launch_bounds__(512) void ripple_finalize(
    const float* __restrict__ base_arr, const float* __restrict__ accum,
    float* __restrict__ out) {
  __shared__ float sb[512];
  const int tid = threadIdx.x;
  sb[tid] = base_arr[tid];
  __syncthreads();
  for (int off = 256; off > 0; off >>= 1) {
    if (tid < off) sb[tid] += sb[tid + off];
    __syncthreads();
  }
  if (tid == 0) {
    float base = sb[0] * (1.0f / (float)BATCH);
    float kge = -KGE_WEIGHT * accum[0] * (1.0f / (float)(BATCH * N_MEMORY));
    float l2 = L2_WEIGHT * accum[1];
    out[0] = base + kge + l2;
    out[1] = base;
    out[2] = kge;
    out[3] = l2;
  }
}

extern "C" void kernel_launch(void* const* d_in, const int* in_sizes, int n_in,
                              void* d_out, int out_size, void* d_ws, size_t ws_size,
                              hipStream_t stream) {
  const float* entity_emb   = (const float*)d_in[0];
  const float* relation_emb = (const float*)d_in[1];
  const float* W_out        = (const float*)d_in[2];
  const float* b_out        = (const float*)d_in[3];
  const int*   items        = (const int*)d_in[4];
  // d_in[5] = labels (unused by the reference loss)
  const int*   memories_h   = (const int*)d_in[6];
  const int*   memories_r   = (const int*)d_in[7];
  const int*   memories_t   = (const int*)d_in[8];

  float* out    = (float*)d_out;
  float* scores = out + 4;                       // [512, 100000]

  // workspace layout
  float* U        = (float*)d_ws;                         // 512*64
  float* base_arr = U + BATCH * DIM;                      // 512
  float* accum    = base_arr + BATCH;                     // 2 floats

  // 1) user embedding (also zeroes accum)
  ripple_uemb<<<BATCH, 64, 0, stream>>>(entity_emb, memories_h, U, accum);

  // 2) scores GEMM: 32 M-tiles * 3125 N-tiles = 100000 wave-tiles, 4 waves/block
  ripple_scores_wmma<<<25000, 128, 0, stream>>>(U, W_out, b_out, scores);

  // 3) per-batch logsumexp - picked score
  ripple_lse<<<BATCH, 256, 0, stream>>>(scores, items, base_arr);

  // 4) hRt triples: 2*512*32 = 32768 waves, 8 waves/block
  ripple_hrt<<<4096, 256, 0, stream>>>(entity_emb, relation_emb,
                                       memories_h, memories_r, memories_t, accum);

  // 5) scalars
  ripple_finalize<<<1, 512, 0, stream>>>(base_arr, accum, out);
}